// RNN_2422361555402
// MI455X (gfx1250) — compile-verified
//
#include <hip/hip_runtime.h>
#include <hip/hip_bf16.h>
#include <math.h>

// ---------------------------------------------------------------------------
// Problem constants (from the reference):  L=8192, I=256, H=2048, O=1
// ---------------------------------------------------------------------------
#define RNN_L 8192
#define RNN_I 256
#define RNN_H 2048

typedef float v2f __attribute__((ext_vector_type(2)));
typedef float v8f __attribute__((ext_vector_type(8)));

// ===========================================================================
// 1) WMMA f32 GEMM with fused bias:  C[M,N] = A[M,K] @ W[N,K]^T + (b0+b1)[N]
//    - 128x128 tile per WG, 8 waves, each wave = 2x4 grid of 16x16 WMMA tiles
//      driven by V_WMMA_F32_16X16X4_F32.
//    - K staged in chunks of 16 through DOUBLE-BUFFERED LDS filled with
//      GLOBAL_LOAD_ASYNC_TO_LDS_B128 (ASYNCcnt), so the next chunk streams in
//      while the current one feeds the WMMAs. Pad stride 18 -> 9m mod 32 is a
//      permutation -> conflict-free b64 fragment reads.
// ===========================================================================
#define GT_M 128
#define GT_N 128
#define GT_K 16
#define GT_LDS 18   // padded floats per LDS row

__global__ __launch_bounds__(256, 2)
void wmma_gemm_bias(const float* __restrict__ A, const float* __restrict__ W,
                    const float* __restrict__ b0, const float* __restrict__ b1,
                    float* __restrict__ C, int M, int N, int K) {
    __shared__ float As[2][GT_M * GT_LDS];   // 2 x 9216 B
    __shared__ float Ws[2][GT_N * GT_LDS];   // 2 x 9216 B   (36 KB total)

    const int tid  = threadIdx.x;
    const int wave = tid >> 5;
    const int lane = tid & 31;
    const int m0 = blockIdx.y * GT_M;
    const int n0 = blockIdx.x * GT_N;

    // wave tile origin inside the 128x128 block: 4 waves along M, 2 along N
    const int wm = (wave >> 1) * 32;   // 0,32,64,96
    const int wn = (wave & 1) * 64;    // 0,64

    v8f acc[2][4];
    #pragma unroll
    for (int i = 0; i < 2; ++i)
        #pragma unroll
        for (int j = 0; j < 4; ++j)
            #pragma unroll
            for (int e = 0; e < 8; ++e) acc[i][j][e] = 0.0f;

    // fragment addressing (ISA 7.12.2):
    //   A 16x4 f32: lanes 0-15 -> K={0,1}, lanes 16-31 -> K={2,3}; M = lane&15
    //   B 4x16 f32: same K split;                         N = lane&15
    const int lm = lane & 15;
    const int kp = (lane >> 4) * 2;

    // async stage of one K-chunk: 128 rows x 16 floats = 512 float4 per matrix,
    // 2 float4 per thread per matrix, direct global -> LDS (no VGPR round trip)
    auto stage = [&](int k0, int b) {
        #pragma unroll
        for (int i = 0; i < 2; ++i) {
            const int id = tid + i * 256;
            const int r  = id >> 2;            // 0..127
            const int c4 = (id & 3) * 4;       // float4 column
            const unsigned la = (unsigned)(size_t)&As[b][r * GT_LDS + c4];
            const unsigned ga = (unsigned)(((size_t)(m0 + r) * K + k0 + c4) * 4);
            asm volatile("global_load_async_to_lds_b128 %0, %1, %2"
                         :: "v"(la), "v"(ga), "s"(A) : "memory");
            const unsigned lw = (unsigned)(size_t)&Ws[b][r * GT_LDS + c4];
            const unsigned gw = (unsigned)(((size_t)(n0 + r) * K + k0 + c4) * 4);
            asm volatile("global_load_async_to_lds_b128 %0, %1, %2"
                         :: "v"(lw), "v"(gw), "s"(W) : "memory");
        }
    };

    const int nch = K / GT_K;
    stage(0, 0);                                   // prologue: fill buffer 0
    for (int c = 0; c < nch; ++c) {
        const int cur = c & 1;
        asm volatile("s_wait_asynccnt 0x0" ::: "memory");  // my copies landed
        __syncthreads();                                    // everyone's landed
        if (c + 1 < nch) stage((c + 1) * GT_K, cur ^ 1);    // prefetch next chunk

        #pragma unroll
        for (int kk = 0; kk < GT_K; kk += 4) {
            v2f a[2], b[4];
            #pragma unroll
            for (int i = 0; i < 2; ++i)
                a[i] = *(const v2f*)&As[cur][(wm + i * 16 + lm) * GT_LDS + kk + kp];
            #pragma unroll
            for (int j = 0; j < 4; ++j)
                b[j] = *(const v2f*)&Ws[cur][(wn + j * 16 + lm) * GT_LDS + kk + kp];
            #pragma unroll
            for (int i = 0; i < 2; ++i)
                #pragma unroll
                for (int j = 0; j < 4; ++j)
                    acc[i][j] = __builtin_amdgcn_wmma_f32_16x16x4_f32(
                        false, a[i], false, b[j], (short)0, acc[i][j], false, false);
        }
    }

    // epilogue: C/D layout (ISA 7.12.2): row m = vgpr + 8*(lane>>4), col n = lane&15
    const int vrow = (lane >> 4) * 8;
    #pragma unroll
    for (int j = 0; j < 4; ++j) {
        const int ng = n0 + wn + j * 16 + lm;
        const float bias = b0[ng] + b1[ng];
        #pragma unroll
        for (int i = 0; i < 2; ++i) {
            const int mg = m0 + wm + i * 16 + vrow;
            #pragma unroll
            for (int v = 0; v < 8; ++v)
                C[(size_t)(mg + v) * N + ng] = acc[i][j][v] + bias;
        }
    }
}

// ===========================================================================
// 2) Persistent recurrence kernel: h_t = tanh(pre_t + W_hh h_{t-1}), in-place
//    over pre[] so the buffer becomes the layer's output sequence.
//    256 WGs x 256 threads; each WG pins 8 rows of W_hh (64 KB) in LDS for the
//    whole kernel, one wave per row. Per step: read 8 KB h from L2, LDS matvec,
//    wave reduce, tanh, publish 8 new elements, device-wide barrier.
// ===========================================================================
#define R_NWG 256
#define R_TPB 256
#define R_ROWS 8   // RNN_H / R_NWG

__device__ inline void grid_sync(unsigned* cnt, volatile unsigned* gen, unsigned nwg) {
    __threadfence();          // release this WG's h writes to device scope
    __syncthreads();
    if (threadIdx.x == 0) {
        const unsigned g = *gen;
        if (atomicAdd(cnt, 1u) == nwg - 1u) {
            atomicSub(cnt, nwg);          // safe vs. early next-phase arrivals
            __threadfence();
            atomicAdd((unsigned*)gen, 1u);
        } else {
            while (*gen == g) { __builtin_amdgcn_s_sleep(1); }
        }
    }
    __syncthreads();
    __threadfence();          // acquire: don't consume stale L0 lines of hbuf
}

__global__ __launch_bounds__(R_TPB, 1)
void rnn_recurrence(const float* __restrict__ Whh, float* __restrict__ pre,
                    float* __restrict__ hbuf, unsigned* __restrict__ bar, int L) {
    __shared__ float wl[R_ROWS * RNN_H];   // 64 KB weight-stationary slice

    const int tid  = threadIdx.x;
    const int wave = tid >> 5;
    const int lane = tid & 31;
    const int rg   = blockIdx.x * R_ROWS + wave;   // this wave's output row

    // preload this WG's contiguous 8 rows of W_hh into LDS (coalesced float4)
    const float* Wb = Whh + (size_t)blockIdx.x * R_ROWS * RNN_H;
    #pragma unroll
    for (int i = 0; i < 16; ++i) {
        const int fo = (tid + i * R_TPB) * 4;      // 16384 floats total
        *(float4*)&wl[fo] = *(const float4*)&Wb[fo];
    }
    // zero both halves of the double-buffered hidden state (h0 = 0)
    const int gt = blockIdx.x * R_TPB + tid;
    if (gt < 2 * RNN_H) hbuf[gt] = 0.0f;
    __syncthreads();
    grid_sync(bar, bar + 1, R_NWG);

    const float* wrow = &wl[wave * RNN_H];
    for (int t = 0; t < L; ++t) {
        const float* h = hbuf + (t & 1) * RNN_H;
        float acc = 0.0f;
        #pragma unroll
        for (int j = 0; j < 16; ++j) {
            const int c = j * 128 + lane * 4;      // conflict-free b128 LDS reads
            const float4 w4 = *(const float4*)&wrow[c];
            const float4 h4 = *(const float4*)&h[c];
            acc += w4.x * h4.x + w4.y * h4.y + w4.z * h4.z + w4.w * h4.w;
        }
        #pragma unroll
        for (int off = 16; off > 0; off >>= 1)
            acc += __shfl_xor(acc, off, 32);
        if (lane == 0) {
            const float hn = tanhf(pre[(size_t)t * RNN_H + rg] + acc);
            pre[(size_t)t * RNN_H + rg]            = hn;  // layer output sequence
            hbuf[((t + 1) & 1) * RNN_H + rg]       = hn;  // next step's input
        }
        grid_sync(bar, bar + 1, R_NWG);
    }
}

// ===========================================================================
// 3) Final fc + sigmoid on the last hidden state (O = 1)
// ===========================================================================
__global__ void fc_sigmoid(const float* __restrict__ hlast, const float* __restrict__ Wfc,
                           const float* __restrict__ bfc, float* __restrict__ out) {
    __shared__ float red[256];
    float s = 0.0f;
    for (int i = threadIdx.x; i < RNN_H; i += 256) s += hlast[i] * Wfc[i];
    red[threadIdx.x] = s;
    __syncthreads();
    for (int st = 128; st > 0; st >>= 1) {
        if (threadIdx.x < st) red[threadIdx.x] += red[threadIdx.x + st];
        __syncthreads();
    }
    if (threadIdx.x == 0) out[0] = 1.0f / (1.0f + __expf(-(red[0] + bfc[0])));
}

__global__ void init_bar(unsigned* bar) {
    if (threadIdx.x < 2) bar[threadIdx.x] = 0u;
}

// ===========================================================================
// Launch: GEMM(pre0) -> recurrence0 (in place) -> GEMM(pre1) -> recurrence1
//         -> fc. Workspace: 2 x 64 MB sequence buffers + h double buffer + bar.
// ===========================================================================
extern "C" void kernel_launch(void* const* d_in, const int* in_sizes, int n_in,
                              void* d_out, int out_size, void* d_ws, size_t ws_size,
                              hipStream_t stream) {
    const float* x    = (const float*)d_in[0];
    const float* Wih0 = (const float*)d_in[1];
    const float* Whh0 = (const float*)d_in[2];
    const float* bih0 = (const float*)d_in[3];
    const float* bhh0 = (const float*)d_in[4];
    const float* Wih1 = (const float*)d_in[5];
    const float* Whh1 = (const float*)d_in[6];
    const float* bih1 = (const float*)d_in[7];
    const float* bhh1 = (const float*)d_in[8];
    const float* Wfc  = (const float*)d_in[9];
    const float* bfc  = (const float*)d_in[10];

    const size_t seq_elems = (size_t)RNN_L * RNN_H;
    float*    pre0 = (float*)d_ws;             // 64 MB
    float*    pre1 = pre0 + seq_elems;         // 64 MB
    float*    hbuf = pre1 + seq_elems;         // 2*H floats (double buffer)
    unsigned* bar  = (unsigned*)(hbuf + 2 * RNN_H);

    init_bar<<<1, 32, 0, stream>>>(bar);

    dim3 gemm_grid(RNN_H / GT_N, RNN_L / GT_M);   // (16, 64)

    // layer 0: pre0 = x @ W_ih0^T + (b_ih0 + b_hh0), then in-place recurrence
    wmma_gemm_bias<<<gemm_grid, 256, 0, stream>>>(x, Wih0, bih0, bhh0, pre0,
                                                  RNN_L, RNN_H, RNN_I);
    rnn_recurrence<<<R_NWG, R_TPB, 0, stream>>>(Whh0, pre0, hbuf, bar, RNN_L);

    // layer 1: pre1 = seq0 @ W_ih1^T + (b_ih1 + b_hh1), then in-place recurrence
    wmma_gemm_bias<<<gemm_grid, 256, 0, stream>>>(pre0, Wih1, bih1, bhh1, pre1,
                                                  RNN_L, RNN_H, RNN_H);
    rnn_recurrence<<<R_NWG, R_TPB, 0, stream>>>(Whh1, pre1, hbuf, bar, RNN_L);

    // head: sigmoid(h_last . W_fc + b_fc)
    fc_sigmoid<<<1, 256, 0, stream>>>(pre1 + (size_t)(RNN_L - 1) * RNN_H,
                                      Wfc, bfc, (float*)d_out);
}